// LearnablePositionEncoding_33277406609938
// MI455X (gfx1250) — compile-verified
//
#include <hip/hip_runtime.h>
#include <hip/hip_bf16.h>
#include <math.h>

// Problem constants (match reference)
#define Bc   4
#define Hc   16
#define Sc   2048
#define Dc   64
#define BHc  (Bc*Hc)          // 64 rows per position
#define MATN (Dc*Dc)          // 4096 floats per 64x64 matrix
#define LDSTR 66              // padded LDS row stride (even -> float2 aligned, kills bank conflicts)
#define NTHREADS 512          // 16 waves of 32

typedef float v2f __attribute__((ext_vector_type(2)));
typedef float v8f __attribute__((ext_vector_type(8)));

// ---------------------------------------------------------------------------
// Stage one contiguous 64x64 matrix (row-major, row stride 64) from global
// into padded LDS (row stride LDSTR).  float4 global loads, float2 LDS stores.
// ---------------------------------------------------------------------------
__device__ __forceinline__ void stage_matrix(float* __restrict__ lds,
                                             const float* __restrict__ g,
                                             int tid) {
  const float4* g4 = (const float4*)g;
  for (int idx = tid; idx < MATN / 4; idx += NTHREADS) {
    const int r  = idx >> 4;          // row 0..63
    const int c4 = (idx & 15) << 2;   // col 0,4,...,60
    const float4 v = g4[idx];
    v2f* p = (v2f*)(lds + r * LDSTR + c4);   // even offset -> 8B aligned
    p[0] = v2f{v.x, v.y};
    p[1] = v2f{v.z, v.w};
  }
}

// ---------------------------------------------------------------------------
// One wave computes one 16x16 tile of C = A(64x64) @ B(64x64), A/B in LDS
// (row stride LDSTR), using V_WMMA_F32_16X16X4_F32 over 16 K-steps.
// A operand layout (ISA 7.12.2, 32-bit A 16x4): lanes 0-15 hold M=lane,
// VGPR0/1 = K0/K1; lanes 16-31 hold K2/K3.  B/C/D: row striped across lanes.
// ---------------------------------------------------------------------------
__device__ __forceinline__ v8f wave_mm64(const float* __restrict__ A,
                                         const float* __restrict__ Bm,
                                         int tm, int tn, int lane) {
  v8f c = {0.f, 0.f, 0.f, 0.f, 0.f, 0.f, 0.f, 0.f};
  const int half = lane >> 4;           // 0: K+0/K+1, 1: K+2/K+3
  const int l15  = lane & 15;
  const int arow = (tm << 4) + l15;     // M index
  const int bcol = (tn << 4) + l15;     // N index
#pragma unroll
  for (int k = 0; k < 16; ++k) {
    const int kb = (k << 2) + (half << 1);
    v2f a, b;
    a = *(const v2f*)(A + arow * LDSTR + kb);   // aligned float2
    b.x = Bm[kb * LDSTR + bcol];
    b.y = Bm[(kb + 1) * LDSTR + bcol];
    c = __builtin_amdgcn_wmma_f32_16x16x4_f32(false, a, false, b,
                                              (short)0, c, false, false);
  }
  return c;
}

__device__ __forceinline__ void store_tile_lds(float* __restrict__ dst, v8f c,
                                               int tm, int tn, int lane) {
  const int half = lane >> 4;
  const int l15  = lane & 15;
#pragma unroll
  for (int v = 0; v < 8; ++v)
    dst[((tm << 4) + v + (half << 3)) * LDSTR + (tn << 4) + l15] = c[v];
}

__device__ __forceinline__ void store_tile_global(float* __restrict__ dst, v8f c,
                                                  int tm, int tn, int lane) {
  const int half = lane >> 4;
  const int l15  = lane & 15;
#pragma unroll
  for (int v = 0; v < 8; ++v)
    dst[((tm << 4) + v + (half << 3)) * Dc + (tn << 4) + l15] = c[v];
}

// ---------------------------------------------------------------------------
// Kernel 1: sequential power chain.  Pow[j] = M^j for j=1..64, and base
// powers Pow[64c] = M^(64c) for c=2..31.  Single block, 16 waves, LDS
// ping-pong; ~93 dependent 64^3 WMMA matmuls (negligible time).
// ---------------------------------------------------------------------------
__global__ __launch_bounds__(NTHREADS)
void lpe_pow_chain(const float* __restrict__ M, float* __restrict__ Pow) {
  __shared__ float Mb[Dc * LDSTR];
  __shared__ float P64[Dc * LDSTR];
  __shared__ float Abuf[Dc * LDSTR];
  __shared__ float Bbuf[Dc * LDSTR];

  const int tid  = threadIdx.x;
  const int lane = tid & 31;
  const int wid  = tid >> 5;            // 0..15
  const int tm   = wid >> 2;
  const int tn   = wid & 3;

  stage_matrix(Mb, M, tid);
  for (int idx = tid; idx < MATN; idx += NTHREADS) {
    const int r = idx >> 6, cc = idx & 63;
    Abuf[r * LDSTR + cc] = M[idx];
    Pow[MATN + idx]      = M[idx];      // Pow[1] = M
  }
  __syncthreads();

  float* cur = Abuf;
  float* nxt = Bbuf;

  // M^2 .. M^64
  for (int j = 2; j <= 64; ++j) {
    v8f acc = wave_mm64(cur, Mb, tm, tn, lane);
    store_tile_lds(nxt, acc, tm, tn, lane);
    store_tile_global(Pow + (size_t)j * MATN, acc, tm, tn, lane);
    __syncthreads();
    float* t = cur; cur = nxt; nxt = t;
  }

  // keep M^64 resident
  for (int idx = tid; idx < Dc * LDSTR; idx += NTHREADS) P64[idx] = cur[idx];
  __syncthreads();

  // M^(64c) for c = 2..31
  for (int c2 = 2; c2 <= 31; ++c2) {
    v8f acc = wave_mm64(cur, P64, tm, tn, lane);
    store_tile_lds(nxt, acc, tm, tn, lane);
    store_tile_global(Pow + (size_t)(c2 * 64) * MATN, acc, tm, tn, lane);
    __syncthreads();
    float* t = cur; cur = nxt; nxt = t;
  }
}

// ---------------------------------------------------------------------------
// Kernel 2: fill remaining powers in parallel: Pow[64c+j] = Pow[64c] @ Pow[j]
// ---------------------------------------------------------------------------
__global__ __launch_bounds__(NTHREADS)
void lpe_pow_fill(float* __restrict__ Pow) {
  const int pos = blockIdx.x;
  if (pos < 65 || (pos & 63) == 0) return;   // block-uniform exit
  const int c = pos >> 6;                    // 1..31
  const int j = pos & 63;                    // 1..63

  __shared__ float Ab[Dc * LDSTR];
  __shared__ float Bb[Dc * LDSTR];

  const int tid  = threadIdx.x;
  const int lane = tid & 31;
  const int wid  = tid >> 5;
  const int tm   = wid >> 2;
  const int tn   = wid & 3;

  stage_matrix(Ab, Pow + (size_t)(c * 64) * MATN, tid);
  stage_matrix(Bb, Pow + (size_t)j * MATN, tid);
  __syncthreads();

  v8f acc = wave_mm64(Ab, Bb, tm, tn, lane);
  store_tile_global(Pow + (size_t)pos * MATN, acc, tm, tn, lane);
}

// ---------------------------------------------------------------------------
// Kernel 3: per position pos: T = X(64x64 key rows) @ Pow[pos], then
// LayerNorm(GELU(T)) per row, fully coalesced output.  pos==0 copies rows.
// ---------------------------------------------------------------------------
__global__ __launch_bounds__(NTHREADS)
void lpe_transform(const float* __restrict__ key,
                   const float* __restrict__ Pow,
                   const float* __restrict__ gamma,
                   const float* __restrict__ beta,
                   float* __restrict__ outK) {
  const int pos = blockIdx.x;
  const int tid = threadIdx.x;

  if (pos == 0) {
    // 64 rows of 64 floats, float4-vectorized, coalesced
    for (int idx = tid; idx < (BHc * Dc) / 4; idx += NTHREADS) {
      const int r  = idx >> 4;
      const int c4 = idx & 15;
      const float4* src = (const float4*)(key + ((size_t)r * Sc) * Dc);
      float4*       dst = (float4*)(outK + ((size_t)r * Sc) * Dc);
      dst[c4] = src[c4];
    }
    return;
  }

  __shared__ float Xb[BHc * LDSTR];
  __shared__ float Pb[Dc * LDSTR];
  __shared__ float Tb[BHc * LDSTR];

  const int lane = tid & 31;
  const int wid  = tid >> 5;
  const int tm   = wid >> 2;
  const int tn   = wid & 3;

  // Stage Pow[pos] (contiguous) and the 64 key rows (each 256B contiguous).
  stage_matrix(Pb, Pow + (size_t)pos * MATN, tid);
  for (int idx = tid; idx < MATN / 4; idx += NTHREADS) {
    const int r  = idx >> 4;          // bh row
    const int c4 = (idx & 15) << 2;
    const float4 v = *(const float4*)(key + ((size_t)r * Sc + pos) * Dc + c4);
    v2f* p = (v2f*)(Xb + r * LDSTR + c4);
    p[0] = v2f{v.x, v.y};
    p[1] = v2f{v.z, v.w};
  }
  __syncthreads();

  v8f acc = wave_mm64(Xb, Pb, tm, tn, lane);
  store_tile_lds(Tb, acc, tm, tn, lane);
  __syncthreads();

  // GELU(exact erf) + LayerNorm: 8 lanes per row, width-8 shuffle reduction.
  const int row = tid >> 3;        // 0..63 (bh)
  const int sub = tid & 7;         // 0..7
  float vals[8];
  float s = 0.f, s2 = 0.f;
#pragma unroll
  for (int i = 0; i < 8; ++i) {
    const int col = (sub << 3) + i;
    const float x = Tb[row * LDSTR + col];
    const float g = 0.5f * x * (1.0f + erff(x * 0.70710678118654752f));
    vals[i] = g;
    s += g;
    s2 += g * g;
  }
#pragma unroll
  for (int m = 1; m < 8; m <<= 1) {
    s  += __shfl_xor(s,  m, 8);
    s2 += __shfl_xor(s2, m, 8);
  }
  const float mean = s * (1.0f / 64.0f);
  const float var  = s2 * (1.0f / 64.0f) - mean * mean;
  const float rstd = rsqrtf(var + 1e-5f);

  // Normalize back into LDS, then stream out coalesced (one wave = one
  // contiguous 256B row as float2 per lane) instead of stride-8 scatter.
#pragma unroll
  for (int i = 0; i < 8; ++i) {
    const int col = (sub << 3) + i;
    Tb[row * LDSTR + col] = (vals[i] - mean) * rstd * gamma[col] + beta[col];
  }
  __syncthreads();

  for (int idx = tid; idx < MATN / 2; idx += NTHREADS) {
    const int r  = idx >> 5;          // row
    const int c2 = (idx & 31) << 1;   // col pair (even -> aligned)
    const v2f v = *(const v2f*)(Tb + r * LDSTR + c2);
    *(v2f*)(outK + ((size_t)r * Sc + pos) * Dc + c2) = v;
  }
}

// ---------------------------------------------------------------------------
// Kernel 4: query passthrough, float4 vectorized copy
// ---------------------------------------------------------------------------
__global__ __launch_bounds__(256)
void lpe_copy_query(const float4* __restrict__ src, float4* __restrict__ dst, int n4) {
  for (int i = blockIdx.x * 256 + threadIdx.x; i < n4; i += gridDim.x * 256)
    dst[i] = src[i];
}

extern "C" void kernel_launch(void* const* d_in, const int* in_sizes, int n_in,
                              void* d_out, int out_size, void* d_ws, size_t ws_size,
                              hipStream_t stream) {
  const float* query = (const float*)d_in[0];
  const float* key   = (const float*)d_in[1];
  const float* M     = (const float*)d_in[2];
  const float* gamma = (const float*)d_in[3];
  const float* beta  = (const float*)d_in[4];

  float* outQ = (float*)d_out;
  float* outK = (float*)d_out + (size_t)BHc * Sc * Dc;
  float* Pow  = (float*)d_ws;                 // 2048 * 4096 floats = 33.5 MB

  // Phase 1: sequential power chain (tiny)
  lpe_pow_chain<<<1, NTHREADS, 0, stream>>>(M, Pow);
  // Phase 2: parallel fill of remaining powers
  lpe_pow_fill<<<Sc, NTHREADS, 0, stream>>>(Pow);
  // Phase 3: transform key (one block per position)
  lpe_transform<<<Sc, NTHREADS, 0, stream>>>(key, Pow, gamma, beta, outK);
  // Query passthrough (independent)
  const int n4 = (BHc * Sc * Dc) / 4;
  lpe_copy_query<<<2048, 256, 0, stream>>>((const float4*)query, (float4*)outQ, n4);
}